// GGNN_3624952398781
// MI455X (gfx1250) — compile-verified
//
#include <hip/hip_runtime.h>
#include <math.h>

// ---------------------------------------------------------------------------
// GGNN forward on MI455X (gfx1250, wave32).
// All GEMMs use V_WMMA_F32_16X16X4_F32 (fp32 exact path — compute is not the
// bottleneck; the 4 GB of edge scatter traffic is). Fragment layouts per
// CDNA5 ISA 7.12.2:
//   A (16x4 f32, 2 VGPRs): lane l: m=l&15, half=l>>4; a.x=A[m][k+2*half],
//                          a.y=A[m][k+2*half+1]
//   B (4x16 f32, 2 VGPRs): lane l: n=l&15, half=l>>4; b.x=B[k+2*half][n],
//                          b.y=B[k+2*half+1][n]
//   C/D (16x16 f32, 8 VGPRs): d[v]: row = v + 8*half, col = lane&15
// ---------------------------------------------------------------------------

#define N_NODES 100000
#define IN_DIM  500
#define H_DIM   128
#define OUT_DIM 40
#define L_LAYERS 4
#define NUM_EDGES 1000000
#define BN_EPS 1e-5f

typedef __attribute__((ext_vector_type(2))) float v2f;
typedef __attribute__((ext_vector_type(8))) float v8f;

__device__ __forceinline__ v8f wmma_f32_k4(v2f a, v2f b, v8f c) {
  return __builtin_amdgcn_wmma_f32_16x16x4_f32(
      /*neg_a=*/false, a, /*neg_b=*/false, b,
      /*c_mod=*/(short)0, c, /*reuse_a=*/false, /*reuse_b=*/false);
}

// ---------------------------------------------------------------------------
// Kernel 1: h = BN1(x @ lin1_w^T + lin1_b)
// One WG = 16 node rows; 8 waves each own one 16-wide column tile of H=128.
// x slab staged in LDS (16 x 500, padded to 504).
// ---------------------------------------------------------------------------
__global__ __launch_bounds__(256) void ggnn_lin1_bn(
    const float* __restrict__ x, const float* __restrict__ w,
    const float* __restrict__ bias,
    const float* __restrict__ gamma, const float* __restrict__ beta,
    const float* __restrict__ mean,  const float* __restrict__ var,
    float* __restrict__ h) {
  __shared__ float xs[16 * 504];
  const int m0  = blockIdx.x * 16;
  const int tid = threadIdx.x;
  for (int i = tid; i < 16 * IN_DIM; i += 256) {
    int r = i / IN_DIM, c = i - r * IN_DIM;
    xs[r * 504 + c] = x[(size_t)(m0 + r) * IN_DIM + c];
  }
  __syncthreads();

  const int wave = tid >> 5;
  const int lane = tid & 31;
  const int mrow = lane & 15;
  const int half = lane >> 4;
  const int n0   = wave * 16;
  const int col  = n0 + mrow;

  v8f acc = {};
  const float* wrow = w + (size_t)col * IN_DIM;  // lin1_w[col][*] (B = W^T)
  const float* arow = xs + mrow * 504;
  for (int k = 0; k < IN_DIM; k += 4) {
    v2f a, b;
    a.x = arow[k + 2 * half];
    a.y = arow[k + 2 * half + 1];
    b.x = wrow[k + 2 * half];
    b.y = wrow[k + 2 * half + 1];
    acc = wmma_f32_k4(a, b, acc);
  }

  const float sc = gamma[col] * rsqrtf(var[col] + BN_EPS);
  const float sh = beta[col] - mean[col] * sc;
  const float bc = bias[col];
#pragma unroll
  for (int v = 0; v < 8; ++v) {
    int row = m0 + v + 8 * half;
    h[(size_t)row * H_DIM + col] = (acc[v] + bc) * sc + sh;
  }
}

// ---------------------------------------------------------------------------
// Kernel 2: m = h @ conv_w[l]   (B is row-major, no transpose)
// ---------------------------------------------------------------------------
__global__ __launch_bounds__(256) void ggnn_conv(
    const float* __restrict__ h, const float* __restrict__ w,
    float* __restrict__ m) {
  __shared__ float hs[16 * 132];
  const int m0  = blockIdx.x * 16;
  const int tid = threadIdx.x;
  for (int i = tid; i < 16 * H_DIM; i += 256) {
    int r = i >> 7, c = i & 127;
    hs[r * 132 + c] = h[(size_t)(m0 + r) * H_DIM + c];
  }
  __syncthreads();

  const int wave = tid >> 5;
  const int lane = tid & 31;
  const int mrow = lane & 15;
  const int half = lane >> 4;
  const int n0   = wave * 16;

  v8f acc = {};
  const float* arow = hs + mrow * 132;
  for (int k = 0; k < H_DIM; k += 4) {
    v2f a, b;
    a.x = arow[k + 2 * half];
    a.y = arow[k + 2 * half + 1];
    b.x = w[(size_t)(k + 2 * half) * H_DIM + n0 + mrow];
    b.y = w[(size_t)(k + 2 * half + 1) * H_DIM + n0 + mrow];
    acc = wmma_f32_k4(a, b, acc);
  }
#pragma unroll
  for (int v = 0; v < 8; ++v) {
    int row = m0 + v + 8 * half;
    m[(size_t)row * H_DIM + n0 + mrow] = acc[v];
  }
}

// ---------------------------------------------------------------------------
// Kernel 3a: zero agg
// ---------------------------------------------------------------------------
__global__ __launch_bounds__(256) void ggnn_zero(float4* __restrict__ p, int n4) {
  int i = blockIdx.x * 256 + threadIdx.x;
  if (i < n4) p[i] = make_float4(0.f, 0.f, 0.f, 0.f);
}

// ---------------------------------------------------------------------------
// Kernel 3b: agg[dst] += m[src]  (segment_sum over 1M edges)
// 32 threads per edge, float4 per thread: coalesced 512B read + contiguous
// global_atomic_add_f32 scatter.
// ---------------------------------------------------------------------------
__global__ __launch_bounds__(256) void ggnn_scatter(
    const float* __restrict__ m, const int* __restrict__ src,
    const int* __restrict__ dst, float* __restrict__ agg) {
  int g = blockIdx.x * 256 + threadIdx.x;          // E*32 total
  int e  = g >> 5;
  int c4 = (g & 31) << 2;
  if (e >= NUM_EDGES) return;
  int s = src[e];
  int d = dst[e];
  const float4 v = *(const float4*)(m + (size_t)s * H_DIM + c4);
  float* out = agg + (size_t)d * H_DIM + c4;
  atomicAdd(out + 0, v.x);
  atomicAdd(out + 1, v.y);
  atomicAdd(out + 2, v.z);
  atomicAdd(out + 3, v.w);
}

// ---------------------------------------------------------------------------
// Kernel 4: fused GRU update (in place on h):
//   gi = agg @ w_ih^T, gh = h @ w_hh^T  (48 WMMA 16x16 tiles staged in LDS)
//   r = sig(gi0+bi0+gh0+bh0); z = sig(gi1+bi1+gh1+bh1)
//   n = tanh(gi2+bi2 + r*(gh2+bh2)); h = (1-z)*n + z*h
// ---------------------------------------------------------------------------
__global__ __launch_bounds__(256) void ggnn_gru(
    const float* __restrict__ agg, const float* __restrict__ w_ih,
    const float* __restrict__ w_hh, const float* __restrict__ b_ih,
    const float* __restrict__ b_hh, float* __restrict__ h) {
  __shared__ float as[16 * 132];        // agg slab
  __shared__ float hs[16 * 132];        // h slab
  __shared__ float gs[2 * 16 * 392];    // gi | gh, 384 cols padded to 392
  const int m0  = blockIdx.x * 16;
  const int tid = threadIdx.x;
  for (int i = tid; i < 16 * H_DIM; i += 256) {
    int r = i >> 7, c = i & 127;
    as[r * 132 + c] = agg[(size_t)(m0 + r) * H_DIM + c];
    hs[r * 132 + c] = h[(size_t)(m0 + r) * H_DIM + c];
  }
  __syncthreads();

  const int wave = tid >> 5;
  const int lane = tid & 31;
  const int mrow = lane & 15;
  const int half = lane >> 4;

#pragma unroll
  for (int i = 0; i < 6; ++i) {
    const int t     = wave * 6 + i;        // 0..47
    const bool isgh = (t >= 24);
    const int n0    = (isgh ? t - 24 : t) * 16;
    const float* slab = isgh ? hs : as;
    const float* W    = isgh ? w_hh : w_ih;
    const float* wrow = W + (size_t)(n0 + mrow) * H_DIM;  // B = W^T
    const float* arow = slab + mrow * 132;
    v8f acc = {};
    for (int k = 0; k < H_DIM; k += 4) {
      v2f a, b;
      a.x = arow[k + 2 * half];
      a.y = arow[k + 2 * half + 1];
      b.x = wrow[k + 2 * half];
      b.y = wrow[k + 2 * half + 1];
      acc = wmma_f32_k4(a, b, acc);
    }
    float* out = gs + (isgh ? 16 * 392 : 0);
#pragma unroll
    for (int v = 0; v < 8; ++v)
      out[(v + 8 * half) * 392 + n0 + mrow] = acc[v];
  }
  __syncthreads();

  for (int idx = tid; idx < 16 * H_DIM; idx += 256) {
    int r = idx >> 7, c = idx & 127;
    const float* gi = gs + r * 392;
    const float* gh = gs + 16 * 392 + r * 392;
    float ir = gi[c]       + b_ih[c];
    float iz = gi[128 + c] + b_ih[128 + c];
    float in = gi[256 + c] + b_ih[256 + c];
    float hr = gh[c]       + b_hh[c];
    float hz = gh[128 + c] + b_hh[128 + c];
    float hn = gh[256 + c] + b_hh[256 + c];
    float rr = 1.f / (1.f + __expf(-(ir + hr)));
    float zz = 1.f / (1.f + __expf(-(iz + hz)));
    float nn = tanhf(in + rr * hn);
    float ho = hs[r * 132 + c];
    h[(size_t)(m0 + r) * H_DIM + c] = (1.f - zz) * nn + zz * ho;
  }
}

// ---------------------------------------------------------------------------
// Kernel 5: emb = BN2(h) @ lin2_w^T + lin2_b   (3 column tiles, cols<40)
// ---------------------------------------------------------------------------
__global__ __launch_bounds__(128) void ggnn_lin2(
    const float* __restrict__ h,
    const float* __restrict__ gamma, const float* __restrict__ beta,
    const float* __restrict__ mean,  const float* __restrict__ var,
    const float* __restrict__ w2, const float* __restrict__ b2,
    float* __restrict__ emb) {
  __shared__ float hs[16 * 132];
  const int m0  = blockIdx.x * 16;
  const int tid = threadIdx.x;
  for (int i = tid; i < 16 * H_DIM; i += 128) {
    int r = i >> 7, c = i & 127;
    float sc = gamma[c] * rsqrtf(var[c] + BN_EPS);
    hs[r * 132 + c] = (h[(size_t)(m0 + r) * H_DIM + c] - mean[c]) * sc + beta[c];
  }
  __syncthreads();

  const int wave = tid >> 5;
  if (wave >= 3) return;   // 48 cols cover OUT=40
  const int lane = tid & 31;
  const int mrow = lane & 15;
  const int half = lane >> 4;
  const int col  = wave * 16 + mrow;
  const bool live = (col < OUT_DIM);

  v8f acc = {};
  const float* wrow = w2 + (size_t)(live ? col : 0) * H_DIM;
  const float* arow = hs + mrow * 132;
  for (int k = 0; k < H_DIM; k += 4) {
    v2f a, b;
    a.x = arow[k + 2 * half];
    a.y = arow[k + 2 * half + 1];
    b.x = wrow[k + 2 * half];
    b.y = wrow[k + 2 * half + 1];
    acc = wmma_f32_k4(a, b, acc);
  }
  if (live) {
    const float bc = b2[col];
#pragma unroll
    for (int v = 0; v < 8; ++v) {
      int row = m0 + v + 8 * half;
      emb[(size_t)row * OUT_DIM + col] = acc[v] + bc;
    }
  }
}

// ---------------------------------------------------------------------------
// Kernel 6: log_softmax per node (40 classes, serial per thread — tiny)
// ---------------------------------------------------------------------------
__global__ __launch_bounds__(256) void ggnn_lsm(
    const float* __restrict__ emb, float* __restrict__ out) {
  int n = blockIdx.x * 256 + threadIdx.x;
  if (n >= N_NODES) return;
  const float* e = emb + (size_t)n * OUT_DIM;
  float mx = e[0];
#pragma unroll
  for (int o = 1; o < OUT_DIM; ++o) mx = fmaxf(mx, e[o]);
  float s = 0.f;
#pragma unroll
  for (int o = 0; o < OUT_DIM; ++o) s += __expf(e[o] - mx);
  float lse = mx + __logf(s);
  float* d = out + (size_t)n * OUT_DIM;
#pragma unroll
  for (int o = 0; o < OUT_DIM; ++o) d[o] = e[o] - lse;
}

// ---------------------------------------------------------------------------
extern "C" void kernel_launch(void* const* d_in, const int* in_sizes, int n_in,
                              void* d_out, int out_size, void* d_ws, size_t ws_size,
                              hipStream_t stream) {
  const float* x        = (const float*)d_in[0];
  const int*   eidx     = (const int*)d_in[1];     // [2, E]: row 0 = src, row 1 = dst
  const float* lin1_w   = (const float*)d_in[2];
  const float* lin1_b   = (const float*)d_in[3];
  const float* bn1_g    = (const float*)d_in[4];
  const float* bn1_b    = (const float*)d_in[5];
  const float* bn1_m    = (const float*)d_in[6];
  const float* bn1_v    = (const float*)d_in[7];
  const float* conv_w   = (const float*)d_in[8];   // [L, H, H]
  const float* gru_w_ih = (const float*)d_in[9];
  const float* gru_w_hh = (const float*)d_in[10];
  const float* gru_b_ih = (const float*)d_in[11];
  const float* gru_b_hh = (const float*)d_in[12];
  const float* bn2_g    = (const float*)d_in[13];
  const float* bn2_b    = (const float*)d_in[14];
  const float* bn2_m    = (const float*)d_in[15];
  const float* bn2_v    = (const float*)d_in[16];
  const float* lin2_w   = (const float*)d_in[17];
  const float* lin2_b   = (const float*)d_in[18];

  const int* src = eidx;
  const int* dst = eidx + NUM_EDGES;

  float* h   = (float*)d_ws;
  float* m   = h + (size_t)N_NODES * H_DIM;
  float* agg = m + (size_t)N_NODES * H_DIM;

  float* log_probs = (float*)d_out;
  float* emb       = log_probs + (size_t)N_NODES * OUT_DIM;

  const int ntiles = N_NODES / 16;  // 6250

  ggnn_lin1_bn<<<ntiles, 256, 0, stream>>>(x, lin1_w, lin1_b, bn1_g, bn1_b,
                                           bn1_m, bn1_v, h);

  const int nh4 = N_NODES * H_DIM / 4;
  const int scatter_blocks = (NUM_EDGES * 32) / 256;
  for (int l = 0; l < L_LAYERS; ++l) {
    ggnn_conv<<<ntiles, 256, 0, stream>>>(h, conv_w + (size_t)l * H_DIM * H_DIM, m);
    ggnn_zero<<<(nh4 + 255) / 256, 256, 0, stream>>>((float4*)agg, nh4);
    ggnn_scatter<<<scatter_blocks, 256, 0, stream>>>(m, src, dst, agg);
    ggnn_gru<<<ntiles, 256, 0, stream>>>(agg, gru_w_ih, gru_w_hh,
                                         gru_b_ih, gru_b_hh, h);
  }

  ggnn_lin2<<<ntiles, 128, 0, stream>>>(h, bn2_g, bn2_b, bn2_m, bn2_v,
                                        lin2_w, lin2_b, emb);
  ggnn_lsm<<<(N_NODES + 255) / 256, 256, 0, stream>>>(emb, log_probs);
}